// PPool3D_25821343383771
// MI455X (gfx1250) — compile-verified
//
#include <hip/hip_runtime.h>

typedef float v2f __attribute__((ext_vector_type(2)));
typedef float v8f __attribute__((ext_vector_type(8)));

// ---------------------------------------------------------------------------
// Kernel 1: exclusive prefix scan of segment sizes -> starts[0..n], starts[n]=total
// Single workgroup, 1024 threads, 16 elements per thread (n_seg <= 16384).
// ---------------------------------------------------------------------------
#define SCAN_THREADS 1024
#define SCAN_PER 16

__global__ __launch_bounds__(SCAN_THREADS)
void seg_exscan_kernel(const int* __restrict__ sizes, int* __restrict__ starts, int n) {
    __shared__ int partial[SCAN_THREADS];
    const int t = threadIdx.x;
    int local[SCAN_PER];
    int sum = 0;
#pragma unroll
    for (int i = 0; i < SCAN_PER; ++i) {
        int idx = t * SCAN_PER + i;
        int v = (idx < n) ? sizes[idx] : 0;
        local[i] = sum;          // exclusive within this thread's chunk
        sum += v;
    }
    partial[t] = sum;
    __syncthreads();
    // Hillis-Steele inclusive scan over the 1024 per-thread totals
    for (int off = 1; off < SCAN_THREADS; off <<= 1) {
        int v = (t >= off) ? partial[t - off] : 0;
        __syncthreads();
        partial[t] += v;
        __syncthreads();
    }
    const int base = (t == 0) ? 0 : partial[t - 1];
#pragma unroll
    for (int i = 0; i < SCAN_PER; ++i) {
        int idx = t * SCAN_PER + i;
        if (idx < n) starts[idx] = base + local[i];
    }
    if (t == SCAN_THREADS - 1) starts[n] = partial[SCAN_THREADS - 1];
}

// ---------------------------------------------------------------------------
// Kernel 2: segment mean as  out = W x data  via V_WMMA_F32_16X16X4_F32.
//   Block = 8 waves (256 threads). Block b owns segments [16b, 16b+16),
//   wave w owns features [16w, 16w+16). The wave streams the tile's row range
//   in chunks of 4 rows; 8 chunks (32 rows) per unrolled iteration so 16 loads
//   are in flight before the WMMA chain consumes them.
//     A (16x4, 2 VGPRs): selector weights from prefix-sum boundaries
//       VGPR0 -> K = (lane<16 ? 0 : 2), VGPR1 -> K+1, M = lane&15
//     B (4x16,  2 VGPRs): 4 data rows x 16 features
//       VGPR0 -> K = (lane<16 ? 0 : 2), VGPR1 -> K+1, N = lane&15
//     C/D (16x16, 8 VGPRs): VGPR j -> row M=j (lanes 0-15) / M=j+8 (lanes 16-31)
//   EXEC stays all-ones (uniform loop bounds). STRIDE template = compile-time
//   feature stride (0 = runtime) so inner-loop addresses are immediate offsets.
// ---------------------------------------------------------------------------
template <int STRIDE>
__global__ __launch_bounds__(256)
void seg_mean_wmma_kernel(const float* __restrict__ data,
                          const int* __restrict__ starts,
                          float* __restrict__ out,
                          int n_seg, int d_feat_rt) {
    const int d_feat = STRIDE ? STRIDE : d_feat_rt;

    const int lane = threadIdx.x & 31;
    const int wave = threadIdx.x >> 5;           // 0..7
    const int s0   = blockIdx.x * 16;            // first segment of this tile
    const int featBase = wave * 16;

    const int n    = lane & 15;                  // B column == A row (M) == store column
    const int kOff = (lane >> 4) << 1;           // 0 for lanes 0-15, 2 for lanes 16-31

    // Per-lane segment parameters (indices clamped so EXEC stays all-ones)
    const int segIdx     = min(s0 + n, n_seg - 1);
    const int segStart   = starts[segIdx];
    const unsigned uSize = (unsigned)(starts[segIdx + 1] - segStart);
    const float invSz    = uSize ? (1.0f / (float)uSize) : 0.0f;

    // Uniform (per-block) row range covered by this 16-segment tile
    const int rowBeg = starts[s0];
    const int rowEnd = starts[min(s0 + 16, n_seg)];

    // Per-lane column pointer: row r lives at lanePtr + (r - kOff)*d_feat
    const float* lanePtr = data + (size_t)featBase + (size_t)n + (size_t)kOff * d_feat;

    v8f acc = {};  // 16x16 f32 accumulator

    int base = rowBeg;
    const int mainEnd = rowBeg + ((rowEnd - rowBeg) & ~31);   // whole 32-row groups

    // ---- main loop: 32 rows (8 chunks, 16 loads, 8 WMMAs) per iteration ----
    for (; base < mainEnd; base += 32) {
        v2f b[8];
        v2f a[8];
#pragma unroll
        for (int u = 0; u < 8; ++u) {
            const float* p = lanePtr + (size_t)(base + 4 * u) * d_feat;
            b[u].x = p[0];
            b[u].y = p[d_feat];
            // Pull the stream 64 rows ahead into L2 (global_prefetch_b8).
            __builtin_prefetch(p + (size_t)64 * d_feat, 0, 1);
            const int r0 = base + 4 * u + kOff;
            a[u].x = ((unsigned)(r0 - segStart) < uSize) ? invSz : 0.0f;
            a[u].y = ((unsigned)(r0 + 1 - segStart) < uSize) ? invSz : 0.0f;
        }
#pragma unroll
        for (int u = 0; u < 8; ++u) {
            acc = __builtin_amdgcn_wmma_f32_16x16x4_f32(
                      false, a[u], false, b[u], (short)0, acc, false, false);
        }
    }

    // ---- tail: single clamped chunks (0..8 iterations) ----
    for (; base < rowEnd; base += 4) {
        const int r0  = base + kOff;
        const int cr0 = (r0 < rowEnd) ? r0 : (rowEnd - 1);
        const int cr1 = (r0 + 1 < rowEnd) ? (r0 + 1) : (rowEnd - 1);
        v2f b;
        b.x = lanePtr[(size_t)(cr0 - kOff) * d_feat + (size_t)kOff * 0];  // see note below
        b.y = lanePtr[(size_t)(cr1 - kOff) * d_feat];
        // lanePtr already folds kOff*d_feat, so index with (row - kOff)
        b.x = lanePtr[(size_t)(cr0 - kOff) * d_feat];
        v2f a;
        a.x = ((unsigned)(r0 - segStart) < uSize) ? invSz : 0.0f;
        a.y = ((unsigned)(r0 + 1 - segStart) < uSize) ? invSz : 0.0f;
        acc = __builtin_amdgcn_wmma_f32_16x16x4_f32(
                  false, a, false, b, (short)0, acc, false, false);
    }

    // ---- store the 16x16 mean tile ----
    const int M0 = (lane >> 4) * 8;      // lanes 16-31 hold rows 8..15
    if (s0 + 16 <= n_seg) {              // uniform fast path: no guards
        float* op = out + (size_t)(s0 + M0) * d_feat + featBase + n;
#pragma unroll
        for (int j = 0; j < 8; ++j) op[(size_t)j * d_feat] = acc[j];
    } else {
#pragma unroll
        for (int j = 0; j < 8; ++j) {
            const int segRow = s0 + M0 + j;
            if (segRow < n_seg)
                out[(size_t)segRow * d_feat + featBase + n] = acc[j];
        }
    }
}

// ---------------------------------------------------------------------------
// Host-side launch
// ---------------------------------------------------------------------------
extern "C" void kernel_launch(void* const* d_in, const int* in_sizes, int n_in,
                              void* d_out, int out_size, void* d_ws, size_t ws_size,
                              hipStream_t stream) {
    (void)n_in; (void)ws_size;
    const float* data     = (const float*)d_in[0];
    const int*   obj_size = (const int*)d_in[1];
    float*       out      = (float*)d_out;

    const int n_seg  = in_sizes[1];            // 16384
    const int d_feat = out_size / n_seg;       // 128

    int* starts = (int*)d_ws;                  // (n_seg + 1) ints of scratch

    // 1) prefix-sum segment boundaries
    seg_exscan_kernel<<<dim3(1), dim3(SCAN_THREADS), 0, stream>>>(obj_size, starts, n_seg);

    // 2) WMMA segment mean: one block per 16 segments, one wave per 16 features
    const int segTiles = (n_seg + 15) / 16;        // 1024
    const int waves    = d_feat / 16;              // 8
    if (d_feat == 128) {
        seg_mean_wmma_kernel<128><<<dim3(segTiles), dim3(waves * 32), 0, stream>>>(
            data, starts, out, n_seg, d_feat);
    } else {
        seg_mean_wmma_kernel<0><<<dim3(segTiles), dim3(waves * 32), 0, stream>>>(
            data, starts, out, n_seg, d_feat);
    }
}